// K_WTA1D_6425271075427
// MI455X (gfx1250) — compile-verified
//
#include <hip/hip_runtime.h>
#include <stdint.h>

// Problem geometry (fixed by the reference): x is [64, 256, 56, 56] f32.
#define NB    64
#define ROW   802816            // 256*56*56 elements per batch row
#define ROWV  (ROW / 4)         // 200704 float4 per row
#define BPR   98                // blocks per row
#define TPB   256
#define ITERS 8                 // 98 * 256 * 8 == 200704 float4  (exact cover)
#define BINS1 4096              // top-12-bit radix histogram
#define CAP   32768             // candidate buffer capacity per row

typedef __attribute__((ext_vector_type(4))) float f4;

// Order-preserving float -> uint32 map (strictly monotonic).
__device__ __forceinline__ unsigned ordu(float f) {
  unsigned u = __float_as_uint(f);
  return (u & 0x80000000u) ? ~u : (u | 0x80000000u);
}

__global__ void zero_kernel(unsigned* __restrict__ p, int n) {
  int i = blockIdx.x * blockDim.x + threadIdx.x;
  if (i < n) p[i] = 0u;
}

// ---------------------------------------------------------------------------
// Pass 1: per-row histogram of top 12 ordered bits.
// Global -> LDS via the CDNA5 async data-mover path (double buffered), then
// ds_load_b128 + LDS atomics.  Default (RT) caching so x stays hot in L2 for
// the compaction pass.
// ---------------------------------------------------------------------------
__global__ void hist_kernel(const float* __restrict__ x,
                            unsigned* __restrict__ ghist) {
  __shared__ unsigned hist[BINS1];
  __shared__ __align__(16) float tile[2][TPB * 4];

  const int tid = threadIdx.x;
  const int row = blockIdx.y;
  const int blk = blockIdx.x;

  for (int i = tid; i < BINS1; i += TPB) hist[i] = 0u;
  __syncthreads();

  const f4* xv      = (const f4*)(x + (size_t)row * ROW);
  const size_t base = (size_t)blk * (TPB * ITERS);

  // Wave-relative LDS byte addresses of this lane's tile slots
  // (low 32 bits of the flat shared-aperture address == LDS offset).
  unsigned lds0 = (unsigned)(uintptr_t)&tile[0][tid * 4];
  unsigned lds1 = (unsigned)(uintptr_t)&tile[1][tid * 4];

  // Prologue: issue async loads for iterations 0 and 1.
  {
    const f4* g0 = xv + base + 0 * TPB + tid;
    asm volatile("global_load_async_to_lds_b128 %0, %1, off"
                 :: "v"(lds0), "v"(g0) : "memory");
    const f4* g1 = xv + base + 1 * TPB + tid;
    asm volatile("global_load_async_to_lds_b128 %0, %1, off"
                 :: "v"(lds1), "v"(g1) : "memory");
  }

  for (int it = 0; it < ITERS; ++it) {
    // Retire the oldest outstanding async load (they complete in order).
    if (it + 1 < ITERS) {
      asm volatile("s_wait_asynccnt 0x1" ::: "memory");
    } else {
      asm volatile("s_wait_asynccnt 0x0" ::: "memory");
    }

    f4 v = *(const f4*)&tile[it & 1][tid * 4];   // ds_load_b128
    atomicAdd(&hist[ordu(v.x) >> 20], 1u);
    atomicAdd(&hist[ordu(v.y) >> 20], 1u);
    atomicAdd(&hist[ordu(v.z) >> 20], 1u);
    atomicAdd(&hist[ordu(v.w) >> 20], 1u);

    // Re-target this buffer only after its values were consumed above
    // (the data dependence guarantees the ds_load completed).
    if (it + 2 < ITERS) {
      const f4* g = xv + base + (size_t)(it + 2) * TPB + tid;
      unsigned l = (it & 1) ? lds1 : lds0;
      asm volatile("global_load_async_to_lds_b128 %0, %1, off"
                   :: "v"(l), "v"(g) : "memory");
    }
  }
  __syncthreads();

  unsigned* gh = ghist + (size_t)row * BINS1;
  for (int i = tid; i < BINS1; i += TPB) {
    unsigned c = hist[i];
    if (c) atomicAdd(&gh[i], c);
  }
}

// ---------------------------------------------------------------------------
// Per-row scan of the 4096-bin histogram from the top: pick bucket b holding
// the k-th largest element and residual rank k' within it.
// ---------------------------------------------------------------------------
__global__ void select1_kernel(const unsigned* __restrict__ ghist,
                               unsigned* __restrict__ bucket,
                               unsigned* __restrict__ kprime, int k) {
  int row = blockIdx.x;
  if (threadIdx.x != 0) return;
  const unsigned* h = ghist + (size_t)row * BINS1;
  unsigned cum = 0, b = 0, kp = 1;
  for (int i = BINS1 - 1; i >= 0; --i) {
    unsigned c = h[i];
    if (cum + c >= (unsigned)k) { b = (unsigned)i; kp = (unsigned)k - cum; break; }
    cum += c;
  }
  bucket[row] = b;
  kprime[row] = kp;
}

// ---------------------------------------------------------------------------
// Pass 2: compact candidates (elements whose top-12 ordered bits == b).
// Wave32-aggregated reservation: one global atomicAdd per wave per hit-group
// instead of one per lane.
// ---------------------------------------------------------------------------
__global__ void compact_kernel(const float* __restrict__ x,
                               const unsigned* __restrict__ bucket,
                               unsigned* __restrict__ counts,
                               unsigned* __restrict__ cand) {
  const int tid  = threadIdx.x;
  const int row  = blockIdx.y;
  const int blk  = blockIdx.x;
  const int lane = tid & 31;
  const unsigned b = bucket[row];
  const f4* xv = (const f4*)(x + (size_t)row * ROW);
  unsigned* crow = cand + (size_t)row * CAP;
  const size_t base = (size_t)blk * (TPB * ITERS) + tid;

  for (int it = 0; it < ITERS; ++it) {
    f4 v = xv[base + (size_t)it * TPB];
    unsigned u[4] = {ordu(v.x), ordu(v.y), ordu(v.z), ordu(v.w)};
#pragma unroll
    for (int j = 0; j < 4; ++j) {
      bool hit = ((u[j] >> 20) == b);
      unsigned long long m = __ballot(hit);          // wave32: low 32 bits
      unsigned cnt = (unsigned)__popcll(m);
      if (cnt) {
        int leader = (int)(__ffsll((unsigned long long)m) - 1);
        unsigned wbase = 0;
        if (lane == leader) wbase = atomicAdd(&counts[row], cnt);
        wbase = (unsigned)__shfl((int)wbase, leader, 32);
        if (hit) {
          unsigned prefix =
              (unsigned)__popcll(m & ((1ull << (unsigned)lane) - 1ull));
          unsigned idx = wbase + prefix;
          if (idx < CAP) crow[idx] = u[j];
        }
      }
    }
  }
}

// ---------------------------------------------------------------------------
// Refine on candidates only: bits 19:8 (4096 bins), then bits 7:0 (256 bins).
// Produces the exact k-th-largest ordered-uint threshold per row.
// ---------------------------------------------------------------------------
__global__ void select2_kernel(const unsigned* __restrict__ cand,
                               const unsigned* __restrict__ counts,
                               const unsigned* __restrict__ bucket,
                               const unsigned* __restrict__ kprime,
                               unsigned* __restrict__ thr) {
  __shared__ unsigned h[BINS1];
  __shared__ unsigned sb2, skpp;
  const int row = blockIdx.x;
  const int tid = threadIdx.x;
  unsigned n = counts[row]; if (n > CAP) n = CAP;
  const unsigned kp = kprime[row];
  const unsigned* c = cand + (size_t)row * CAP;

  for (int i = tid; i < BINS1; i += TPB) h[i] = 0u;
  __syncthreads();
  for (unsigned i = tid; i < n; i += TPB)
    atomicAdd(&h[(c[i] >> 8) & 0xFFFu], 1u);
  __syncthreads();
  if (tid == 0) {
    unsigned cum = 0, b2 = 0, kpp = 1;
    for (int i = BINS1 - 1; i >= 0; --i) {
      unsigned cc = h[i];
      if (cum + cc >= kp) { b2 = (unsigned)i; kpp = kp - cum; break; }
      cum += cc;
    }
    sb2 = b2; skpp = kpp;
  }
  __syncthreads();
  const unsigned b2 = sb2, kpp = skpp;

  for (int i = tid; i < 256; i += TPB) h[i] = 0u;
  __syncthreads();
  for (unsigned i = tid; i < n; i += TPB) {
    unsigned u = c[i];
    if (((u >> 8) & 0xFFFu) == b2) atomicAdd(&h[u & 0xFFu], 1u);
  }
  __syncthreads();
  if (tid == 0) {
    unsigned cum = 0, b3 = 0;
    for (int i = 255; i >= 0; --i) {
      unsigned cc = h[i];
      if (cum + cc >= kpp) { b3 = (unsigned)i; break; }
      cum += cc;
    }
    thr[row] = (bucket[row] << 20) | (b2 << 8) | b3;
  }
}

// ---------------------------------------------------------------------------
// Pass 3: streaming mask, compare in ordered-uint domain (strict <).
// Non-temporal load (last use of x) and non-temporal store (out is never
// re-read) so the streaming traffic doesn't churn the 192MB L2.
// ---------------------------------------------------------------------------
__global__ void mask_kernel(const float* __restrict__ x,
                            const unsigned* __restrict__ thr,
                            float* __restrict__ out) {
  const int tid = threadIdx.x;
  const int row = blockIdx.y;
  const int blk = blockIdx.x;
  const unsigned t = thr[row];
  const f4* xv = (const f4*)(x + (size_t)row * ROW);
  f4* ov = (f4*)(out + (size_t)row * ROW);
  const size_t base = (size_t)blk * (TPB * ITERS) + tid;

  for (int it = 0; it < ITERS; ++it) {
    size_t i = base + (size_t)it * TPB;
    f4 v = __builtin_nontemporal_load(&xv[i]);
    f4 r;
    r.x = (ordu(v.x) < t) ? v.x : 0.0f;
    r.y = (ordu(v.y) < t) ? v.y : 0.0f;
    r.z = (ordu(v.z) < t) ? v.z : 0.0f;
    r.w = (ordu(v.w) < t) ? v.w : 0.0f;
    __builtin_nontemporal_store(r, &ov[i]);
  }
}

extern "C" void kernel_launch(void* const* d_in, const int* in_sizes, int n_in,
                              void* d_out, int out_size, void* d_ws, size_t ws_size,
                              hipStream_t stream) {
  (void)in_sizes; (void)n_in; (void)out_size; (void)ws_size;
  const float* x = (const float*)d_in[0];
  float* out = (float*)d_out;

  char* ws = (char*)d_ws;
  unsigned* ghist  = (unsigned*)(ws);                       // 64*4096 u32 = 1 MB
  unsigned* bucket = (unsigned*)(ws + 1048576);             // 64 u32
  unsigned* kprime = (unsigned*)(ws + 1048576 + 256);       // 64 u32
  unsigned* counts = (unsigned*)(ws + 1048576 + 512);       // 64 u32
  unsigned* thr    = (unsigned*)(ws + 1048576 + 768);       // 64 u32
  unsigned* cand   = (unsigned*)(ws + 1048576 + 4096);      // 64*32768 u32 = 8 MB

  const int k = (int)(0.1 * (double)ROW);                   // 80281 == int(GAMMA*layer)

  const int zero_n = NB * BINS1 + 1024;                     // hist + header (incl. counts)
  zero_kernel<<<(zero_n + TPB - 1) / TPB, TPB, 0, stream>>>(ghist, zero_n);

  dim3 grid(BPR, NB);
  hist_kernel<<<grid, TPB, 0, stream>>>(x, ghist);
  select1_kernel<<<NB, 64, 0, stream>>>(ghist, bucket, kprime, k);
  compact_kernel<<<grid, TPB, 0, stream>>>(x, bucket, counts, cand);
  select2_kernel<<<NB, TPB, 0, stream>>>(cand, counts, bucket, kprime, thr);
  mask_kernel<<<grid, TPB, 0, stream>>>(x, thr, out);
}